// ocnn_13761075216585
// MI455X (gfx1250) — compile-verified
//
#include <hip/hip_runtime.h>
#include <math.h>

#define EPSF 1e-8f

typedef float v2f __attribute__((ext_vector_type(2)));
typedef float v8f __attribute__((ext_vector_type(8)));

// ---------------- workspace layout (floats) ----------------
// Batch-independent constants, computed once per launch by ocnn_setup.
#define OFF_EIR   0                        // Ei real,  [64][52] row-major (K-contig)
#define OFF_EII   (OFF_EIR + 64*52)        // Ei imag
#define OFF_WTR   (OFF_EII + 64*52)        // DFT W real, transposed [64 cols][200 c] (K-contig)
#define OFF_WTI   (OFF_WTR + 64*200)       // DFT W imag, transposed
#define OFF_WTIN  (OFF_WTI + 64*200)       // -W imag, transposed
#define OFF_WN    (OFF_WTIN + 64*200)      // wn[j][i] = |w[i,j]|/(wmax_i+eps), [50][50]
#define OFF_WMAX  (OFF_WN + 2500)          // wmax[i], 50 (padded 64)
#define OFF_AMPN  (OFF_WMAX + 64)          // pooled amp / (ampmax+eps), [50][50]
#define WS_FLOATS (OFF_AMPN + 2500)        // ~200 KB

static __device__ __forceinline__ v8f wmma_f32(v2f a, v2f b, v8f c) {
  // V_WMMA_F32_16X16X4_F32 : D(16x16,f32) = A(16x4,f32) x B(4x16,f32) + C
  return __builtin_amdgcn_wmma_f32_16x16x4_f32(
      /*neg_a=*/false, a, /*neg_b=*/false, b,
      /*c_mod=*/(short)0, c, /*reuse_a=*/false, /*reuse_b=*/false);
}

static __device__ __forceinline__ v2f ld2(const float* p) {
  return *(const v2f*)p;   // callers guarantee 8B alignment (even float offset)
}

// ================= setup: batch-independent constants =================
__global__ __launch_bounds__(256) void ocnn_setup(
    const float* __restrict__ weight,     // [50,16]
    const float* __restrict__ full_sig,   // [200,200]
    float* __restrict__ ws)
{
  __shared__ float twc[200], tws[200];    // e^{-2pi i m/200}
  __shared__ float wmaxs[50];
  __shared__ float ampf[100 * 52];
  __shared__ float pool[2500];
  __shared__ float red[256];
  const int tid = threadIdx.x;

  for (int m = tid; m < 200; m += 256) {
    float s, c;
    __sincosf(-6.28318530717958647692f * (float)m / 200.0f, &s, &c);
    twc[m] = c; tws[m] = s;
  }
  if (tid < 50) {
    float mx = 0.f;
    for (int j = 0; j < 16; ++j) mx = fmaxf(mx, fabsf(weight[tid * 16 + j]));
    wmaxs[tid] = mx;
    ws[OFF_WMAX + tid] = mx;
  }
  __syncthreads();

  // Ei[i][r] = exp(-2pi i (175+i)(75+r)/200), padded [64][52], K(r)-contiguous
  for (int idx = tid; idx < 64 * 52; idx += 256) {
    int i = idx / 52, r = idx % 52;
    float cr = 0.f, ci = 0.f;
    if (i < 50 && r < 50) {
      int m = ((175 + i) * (75 + r)) % 200;
      cr = twc[m]; ci = tws[m];
    }
    ws[OFF_EIR + idx] = cr;
    ws[OFF_EII + idx] = ci;
  }
  // WT[col][c] = exp(-2pi i (col+1) c / 200), transposed so K(c) is contiguous
  for (int idx = tid; idx < 64 * 200; idx += 256) {
    int col = idx / 200, c = idx % 200;
    float cr = 0.f, ci = 0.f;
    if (col < 52) {
      int m = ((col + 1) * c) % 200;
      cr = twc[m]; ci = tws[m];
    }
    ws[OFF_WTR + idx]  = cr;
    ws[OFF_WTI + idx]  = ci;
    ws[OFF_WTIN + idx] = -ci;
  }
  // wn[j][i]
  for (int idx = tid; idx < 2500; idx += 256) {
    int j = idx / 50, i = idx % 50;
    ws[OFF_WN + idx] = (j < 16) ? (fabsf(weight[i * 16 + j]) / (wmaxs[i] + EPSF)) : 0.f;
  }
  // amp grid: |FFT_axis0(full_sig)| at shifted rows 50..149, cols 101..152
  for (int idx = tid; idx < 100 * 52; idx += 256) {
    int a = idx / 52, cc = idx % 52;
    int k = (150 + a) % 200;
    float re = 0.f, im = 0.f;
    int mm = 0;
    for (int p = 0; p < 200; ++p) {
      float fs = full_sig[p * 200 + 101 + cc];
      re = fmaf(fs, twc[mm], re);
      im = fmaf(fs, tws[mm], im);
      mm += k; if (mm >= 200) mm -= 200;
    }
    ampf[idx] = sqrtf(re * re + im * im);
  }
  __syncthreads();

  // maxpool (2x3, stride 2x1) + global max
  float lm = 0.f;
  for (int idx = tid; idx < 2500; idx += 256) {
    int jj = idx / 50, p = idx % 50;
    float mx = 0.f;
    for (int da = 0; da < 2; ++da)
      for (int t = 0; t < 3; ++t)
        mx = fmaxf(mx, ampf[(2 * jj + da) * 52 + p + t]);
    pool[idx] = mx;
    lm = fmaxf(lm, mx);
  }
  red[tid] = lm;
  __syncthreads();
  for (int s = 128; s > 0; s >>= 1) {
    if (tid < s) red[tid] = fmaxf(red[tid], red[tid + s]);
    __syncthreads();
  }
  float ampmax = red[0];
  for (int idx = tid; idx < 2500; idx += 256)
    ws[OFF_AMPN + idx] = pool[idx] / (ampmax + EPSF);
}

// ================= per-batch pipeline: one workgroup per image =================
__global__ __launch_bounds__(256) void ocnn_main(
    const float* __restrict__ input,      // [B,1,28,28]
    const float* __restrict__ bias,       // [1,49,50]
    const float* __restrict__ full_sig,   // [200,200]
    const float* __restrict__ ws,
    float* __restrict__ out)              // [B,49,50]
{
  // 159.5 KB total static LDS (<160 KB)
  __shared__ __align__(16) float sPT[208 * 52];  // P transposed: [col c][row r], K-contig
  __shared__ __align__(16) float sXR[64 * 200];  // Re X, rows 50..63 exactly zero
  __shared__ __align__(16) float sXI[64 * 200];  // Im X
  __shared__ float sMg[50 * 52];                 // |Y| magnitude grid
  __shared__ float sXn[50 * 16];                 // normalized patch rows
  __shared__ float sImax[64];

  const int tid  = threadIdx.x;
  const int wave = tid >> 5;
  const int lane = tid & 31;
  const int half = lane >> 4;             // lane group 0..15 / 16..31
  const int l15  = lane & 15;
  const int b    = blockIdx.x;
  const float* img = input + b * 784;

  const float* EiR  = ws + OFF_EIR;
  const float* EiI  = ws + OFF_EII;
  const float* WTr  = ws + OFF_WTR;
  const float* WTi  = ws + OFF_WTI;
  const float* WTin = ws + OFF_WTIN;
  const float* WN   = ws + OFF_WN;
  const float* WMX  = ws + OFF_WMAX;
  const float* AMPN = ws + OFF_AMPN;

  // ---- phase 0: 4x4 patches, per-row max, normalize, build P^T
  if (tid < 50) {
    int r = tid;
    float vals[16];
    float mx = 0.f;
    if (r < 49) {
      int pr = r / 7, pc = r % 7;
      for (int j = 0; j < 16; ++j) {
        float v = img[(pr * 4 + (j >> 2)) * 28 + pc * 4 + (j & 3)];
        vals[j] = v;
        mx = fmaxf(mx, v);
      }
    } else {
      for (int j = 0; j < 16; ++j) vals[j] = 0.f;
    }
    sImax[r] = mx;
    float d = mx + EPSF;
    for (int j = 0; j < 16; ++j) sXn[r * 16 + j] = vals[j] / d;
  }
  __syncthreads();
  // PT[c][r] = full_sig[75+r][c] * inpn[r][c/4]; zero for c>=200 (pad), r>=50, c>=64
  for (int idx = tid; idx < 208 * 52; idx += 256) {
    int c = idx / 52, r = idx % 52, j = c >> 2;
    float v = 0.f;
    if (c < 200 && r < 50 && j < 16)
      v = full_sig[(75 + r) * 200 + c] * sXn[r * 16 + j];
    sPT[idx] = v;
  }
  __syncthreads();

  // ---- phase 1: X = Ei @ P (complex via two real WMMA matmuls), M=64,N=208,K=52
  // All fragment loads are unpredicated aligned 64-bit loads.
  for (int job = wave; job < 104; job += 8) {
    int part = job / 52, rem = job % 52;
    int mt = rem / 13, nt = rem % 13;
    const float* E = part ? EiI : EiR;
    float* Xo = part ? sXI : sXR;
    int mB = mt * 16, nB = nt * 16;
    const float* Ea = E + (mB + l15) * 52;         // A row, K-contig
    const float* Pb = &sPT[(nB + l15) * 52];       // B col, K-contig
    v8f acc = {0.f, 0.f, 0.f, 0.f, 0.f, 0.f, 0.f, 0.f};
#pragma unroll
    for (int kt = 0; kt < 13; ++kt) {
      int k0 = kt * 4 + 2 * half;                  // even -> 8B aligned
      acc = wmma_f32(ld2(Ea + k0), ld2(Pb + k0), acc);
    }
    int col = nB + l15;
    if (col < 200) {                               // once per job, outside K loop
#pragma unroll
      for (int v = 0; v < 8; ++v)
        Xo[(mB + v + 8 * half) * 200 + col] = acc[v];   // rows 50..63 store zeros
    }
  }
  __syncthreads();

  // ---- phase 2: apply Bm mask: X''[i,c] = X[i,c] * wn[c/4][i]  (rows >=50 stay 0)
  for (int idx = tid; idx < 50 * 200; idx += 256) {
    int i = idx / 200, c = idx % 200;
    float f = WN[(c >> 2) * 50 + i];
    sXR[idx] *= f;
    sXI[idx] *= f;
  }
  __syncthreads();

  // ---- phase 3: Y = X'' @ W (complex x complex), M=64,N=64,K=200; store |Y|
  for (int job = wave; job < 16; job += 8) {
    int mt = job / 4, nt = job % 4;
    int mB = mt * 16, nB = nt * 16;
    const float* ar  = &sXR[(mB + l15) * 200];     // A rows, K-contig (rows<=63 valid)
    const float* ai  = &sXI[(mB + l15) * 200];
    const float* br  = WTr  + (nB + l15) * 200;    // B cols, K-contig
    const float* bi  = WTi  + (nB + l15) * 200;
    const float* bin = WTin + (nB + l15) * 200;
    v8f cre = {0.f, 0.f, 0.f, 0.f, 0.f, 0.f, 0.f, 0.f};
    v8f cim = {0.f, 0.f, 0.f, 0.f, 0.f, 0.f, 0.f, 0.f};
#pragma unroll 2
    for (int kt = 0; kt < 50; ++kt) {
      int k0 = kt * 4 + 2 * half;
      v2f aR = ld2(ar + k0),  aI = ld2(ai + k0);
      v2f bR = ld2(br + k0),  bI = ld2(bi + k0), bIn = ld2(bin + k0);
      cre = wmma_f32(aR, bR,  cre);   // Re += XR*WR
      cre = wmma_f32(aI, bIn, cre);   // Re -= XI*WI  (pre-negated B)
      cim = wmma_f32(aR, bI,  cim);   // Im += XR*WI
      cim = wmma_f32(aI, bR,  cim);   // Im += XI*WR
    }
#pragma unroll
    for (int v = 0; v < 8; ++v) {
      int row = mB + v + 8 * half;
      int col = nB + l15;
      if (row < 50 && col < 52)
        sMg[row * 52 + col] = sqrtf(cre[v] * cre[v] + cim[v] * cim[v]);
    }
  }
  __syncthreads();

  // ---- phase 4: maxpool(2x3) over zero-padded rows, /amp, per-batch max
  float* res = sPT;          // sPT is dead after phase 1
  float* red = sPT + 2500;
  float lm = 0.f;
  for (int idx = tid; idx < 2500; idx += 256) {
    int jj = idx / 50, p = idx % 50;
    float cm = 0.f;          // rows outside [75,124] of opc_fft are exactly zero
    for (int da = 0; da < 2; ++da) {
      int i = 50 + 2 * jj + da - 75;
      if (i >= 0 && i < 50)
        for (int t = 0; t < 3; ++t) cm = fmaxf(cm, sMg[i * 52 + p + t]);
    }
    float rv = fabsf(cm / AMPN[idx]);
    res[idx] = rv;
    lm = fmaxf(lm, rv);
  }
  red[tid] = lm;
  __syncthreads();
  for (int s = 128; s > 0; s >>= 1) {
    if (tid < s) red[tid] = fmaxf(red[tid], red[tid + s]);
    __syncthreads();
  }
  float rmax = red[0];

  // ---- phase 5: transpose, rescale by imax (x) wmax, add bias
  for (int idx = tid; idx < 49 * 50; idx += 256) {
    int p = idx / 50, q = idx % 50;
    float v = res[q * 50 + p] / (rmax + EPSF) * sImax[p] * WMX[q] + bias[idx];
    out[b * 2450 + idx] = v;
  }
}

// ================= launch =================
extern "C" void kernel_launch(void* const* d_in, const int* in_sizes, int n_in,
                              void* d_out, int out_size, void* d_ws, size_t ws_size,
                              hipStream_t stream) {
  (void)n_in; (void)out_size; (void)ws_size;
  const float* input    = (const float*)d_in[0];
  const float* weight   = (const float*)d_in[1];
  const float* bias     = (const float*)d_in[2];
  const float* full_sig = (const float*)d_in[3];
  float* ws  = (float*)d_ws;   // needs WS_FLOATS*4 ≈ 200 KB
  float* out = (float*)d_out;

  int batch = in_sizes[0] / 784;   // [B,1,28,28]
  ocnn_setup<<<1, 256, 0, stream>>>(weight, full_sig, ws);
  ocnn_main<<<batch, 256, 0, stream>>>(input, bias, full_sig, ws, out);
}